// CrossLayer_54657753809485
// MI455X (gfx1250) — compile-verified
//
#include <hip/hip_runtime.h>

typedef __attribute__((ext_vector_type(2))) float v2f;
typedef __attribute__((ext_vector_type(8))) float v8f;

#define ROWS   16      // rows (examples) per workgroup
#define RSTR   1028    // LDS row stride in floats (1024 + 4 pad -> conflict-free WMMA A loads)
#define NTHR   256     // 8 wave32
#define DIM    1024    // feature dim
#define NLAYER 3

__global__ __launch_bounds__(NTHR) void dcn_cross_wmma_kernel(
    const float* __restrict__ x,
    const float* __restrict__ W,
    const float* __restrict__ bias,
    float* __restrict__ out)
{
    extern __shared__ float smem[];
    float* cr   = smem;               // cross tile: [ROWS][RSTR]
    float* sacc = smem + ROWS * RSTR; // per-layer row sums: [NLAYER][16]

    const int tid  = threadIdx.x;
    const int lane = tid & 31;
    const int wave = tid >> 5;
    const long rowBase = (long)blockIdx.x * ROWS;

    // ---- Load x tile: thread t owns float4 column-block t of every row. ----
    float4 xr[ROWS];
#pragma unroll
    for (int r = 0; r < ROWS; ++r) {
        xr[r] = ((const float4*)(x + (rowBase + r) * DIM))[tid];   // coalesced b128
        ((float4*)(cr + r * RSTR))[tid] = xr[r];                   // cross = x
    }
    if (tid < NLAYER * 16) sacc[tid] = 0.0f;
    __syncthreads();

    // Per-wave K slice: wave w covers K in [w*128, w*128+128), 32 chunks of 4.
    const int kbase = wave * 128;
    const int koff  = (lane >> 4) * 2;   // lanes 0-15: K+0,K+1 ; lanes 16-31: K+2,K+3
    const int arow  = lane & 15;         // A-matrix row M = lane

    for (int i = 0; i < NLAYER; ++i) {
        // ---- s[m] = sum_d cross[m,d] * W[i,d] via V_WMMA_F32_16X16X4_F32 ----
        v8f c = {0.f, 0.f, 0.f, 0.f, 0.f, 0.f, 0.f, 0.f};
        const float* wl = W + i * DIM + kbase + koff;      // B fragment source (broadcast along N)
        const float* al = cr + arow * RSTR + kbase + koff; // A fragment source (LDS, bank-padded)
#pragma unroll
        for (int ch = 0; ch < 32; ++ch) {
            v2f a = *(const v2f*)(al + ch * 4);   // ds_load_b64, 64-bank conflict-free
            v2f b = *(const v2f*)(wl + ch * 4);   // tiny, L2-resident, lane-broadcast
            c = __builtin_amdgcn_wmma_f32_16x16x4_f32(
                    /*neg_a=*/false, a, /*neg_b=*/false, b,
                    /*c_mod=*/(short)0, c, /*reuse_a=*/false, /*reuse_b=*/false);
        }

        // C layout: VGPR j @ lanes 0-15 = C[M=j][N=lane]; @ lanes 16-31 = C[M=j+8][N=lane-16].
        // Columns are identical (B broadcast along N) -> lanes 0 and 16 carry all 16 row sums.
        float* s = sacc + i * 16;
        if ((lane & 15) == 0) {
            const int half = (lane >> 4) * 8;
#pragma unroll
            for (int j = 0; j < 8; ++j) atomicAdd(&s[half + j], c[j]);  // ds_add_f32 x8 waves
        }
        __syncthreads();

        // ---- cross = x * s + bias + cross (streaming update, LDS-resident) ----
        const float4 bi = ((const float4*)(bias + i * DIM))[tid];
#pragma unroll
        for (int r = 0; r < ROWS; ++r) {
            const float sv = s[r];                              // LDS broadcast read
            float4 cv = ((float4*)(cr + r * RSTR))[tid];
            cv.x = xr[r].x * sv + bi.x + cv.x;
            cv.y = xr[r].y * sv + bi.y + cv.y;
            cv.z = xr[r].z * sv + bi.z + cv.z;
            cv.w = xr[r].w * sv + bi.w + cv.w;
            ((float4*)(cr + r * RSTR))[tid] = cv;
            if (i == NLAYER - 1)                                 // fused final store, coalesced b128
                ((float4*)(out + (rowBase + r) * DIM))[tid] = cv;
        }
        __syncthreads();
    }
}

extern "C" void kernel_launch(void* const* d_in, const int* in_sizes, int n_in,
                              void* d_out, int out_size, void* d_ws, size_t ws_size,
                              hipStream_t stream) {
    const float* x    = (const float*)d_in[0];   // (B, D, 1) f32
    const float* W    = (const float*)d_in[1];   // (L, D, 1) f32
    const float* bias = (const float*)d_in[2];   // (L, D, 1) f32
    float* out        = (float*)d_out;           // (B, D, 1) f32

    const int Bn = in_sizes[0] / DIM;            // 16384
    const size_t shmem = (size_t)(ROWS * RSTR + NLAYER * 16) * sizeof(float); // ~64.4 KB

    dim3 grid(Bn / ROWS), block(NTHR);
    dcn_cross_wmma_kernel<<<grid, block, shmem, stream>>>(x, W, bias, out);
}